// RelationMessagePassingLayer_4243427688888
// MI455X (gfx1250) — compile-verified
//
#include <hip/hip_runtime.h>
#include <math.h>

#define NN    1536
#define DD    256
#define HH    4
#define DE_IN 16
#define DEE   8
#define DHH   64
#define JS    8            // j-dimension splits (flash-decoding)
#define JCH   (NN / JS)    // 192 j's per split, 3 chunks of 64
#define NEG_INF (-__builtin_inff())

typedef __attribute__((ext_vector_type(2))) float v2f;
typedef __attribute__((ext_vector_type(8))) float v8f;

__device__ __forceinline__ v8f wmma_f32(v2f a, v2f b, v8f c) {
  // (neg_a, A, neg_b, B, c_mod, C, reuse_a, reuse_b)
  return __builtin_amdgcn_wmma_f32_16x16x4_f32(false, a, false, b, (short)0, c,
                                               false, false);
}

// ---------------------------------------------------------------------------
// Fold We/be through ue:  w_eff[h][c] = sum_de ue[h][de]*We[h*8+de][c]
// ---------------------------------------------------------------------------
__global__ void prep_weff(const float* __restrict__ u, const float* __restrict__ We,
                          const float* __restrict__ be, float* __restrict__ weff,
                          float* __restrict__ beff) {
  int t = threadIdx.x;            // 64 threads
  int h = t >> 4, cc = t & 15;
  const float* ue = u + h * (2 * DHH + DEE) + 2 * DHH;
  float w = 0.f;
  #pragma unroll
  for (int de = 0; de < DEE; ++de) w += ue[de] * We[(size_t)(h * DEE + de) * DE_IN + cc];
  weff[t] = w;
  if (cc == 0) {
    float bb = 0.f;
    #pragma unroll
    for (int de = 0; de < DEE; ++de) bb += ue[de] * be[h * DEE + de];
    beff[h] = bb;
  }
}

// ---------------------------------------------------------------------------
// C[N,256] = A[N,256] @ W[256,256]^T + bias (+ add0 + add1), via f32 WMMA.
// One 16x16 tile per wave, 4 waves per block.
// ---------------------------------------------------------------------------
__global__ __launch_bounds__(128) void gemm_wmma(
    const float* __restrict__ A, const float* __restrict__ W,
    const float* __restrict__ bias, float* __restrict__ out,
    const float* __restrict__ add0, const float* __restrict__ add1) {
  int wave = threadIdx.y;
  int tile = blockIdx.x * 4 + wave;      // N/16 * D/16 = 96*16 tiles
  int tn = tile & 15;                    // D/16 = 16
  int tm = tile >> 4;
  int lane = threadIdx.x;
  int half = lane >> 4;                  // K-half select per f32 A/B layout
  int l    = lane & 15;

  const float* arow = A + (size_t)(tm * 16 + l) * DD;  // A row M=l
  const float* wrow = W + (size_t)(tn * 16 + l) * DD;  // B col N=l (W row-major)
  v8f c = {};
  for (int kb = 0; kb < DD; kb += 4) {
    v2f a, b;
    a.x = arow[kb + half * 2 + 0];
    a.y = arow[kb + half * 2 + 1];
    b.x = wrow[kb + half * 2 + 0];
    b.y = wrow[kb + half * 2 + 1];
    c = wmma_f32(a, b, c);
  }
  int col = tn * 16 + l;
  float bv = bias[col];
  #pragma unroll
  for (int r = 0; r < 8; ++r) {
    int row = tm * 16 + r + half * 8;    // C layout: lanes>=16 hold M=r+8
    float val = c[r] + bv;
    if (add0) val += add0[(size_t)row * DD + col];
    if (add1) val += add1[(size_t)row * DD + col];
    out[(size_t)row * DD + col] = val;
  }
}

// ---------------------------------------------------------------------------
// a[i,h] = q[i,h,:].uq[h] ;  b[i,h] = k[i,h,:].uk[h]
// ---------------------------------------------------------------------------
__global__ __launch_bounds__(256) void qk_dots(const float* __restrict__ q,
                                               const float* __restrict__ k,
                                               const float* __restrict__ u,
                                               float* __restrict__ av,
                                               float* __restrict__ bv) {
  int gid = blockIdx.x * 256 + threadIdx.x;
  if (gid >= NN * HH) return;
  int i = gid >> 2, h = gid & 3;
  const float* qr = q + (size_t)i * DD + h * DHH;
  const float* kr = k + (size_t)i * DD + h * DHH;
  const float* ur = u + h * (2 * DHH + DEE);
  float sa = 0.f, sb = 0.f;
  for (int d = 0; d < DHH; ++d) { sa += qr[d] * ur[d]; sb += kr[d] * ur[DHH + d]; }
  av[gid] = sa;
  bv[gid] = sb;
}

// ---------------------------------------------------------------------------
// Split-j flash attention partial: block = (16 receiver rows) x (192-j slice).
// Grid (96, JS). Emits unnormalized P (at local max scale), plus m, s.
// 8 waves; each wave owns 2 (head, d-tile) tiles via WMMA f32 16x16x4.
// ---------------------------------------------------------------------------
__global__ __launch_bounds__(256) void attn_split(
    const float* __restrict__ adj, const float* __restrict__ ef,
    const float* __restrict__ v,   const float* __restrict__ avec,
    const float* __restrict__ bvec, const float* __restrict__ weff,
    const float* __restrict__ beff, float* __restrict__ Pacc,
    float* __restrict__ m_out, float* __restrict__ s_out) {
  __shared__ float Wl[HH][16][64];    // logits -> exp-weights (A matrices)
  __shared__ float adjl[16][64];
  __shared__ float m_l[16][HH];
  __shared__ float s_l[16][HH];
  __shared__ float sc_l[16][HH];
  __shared__ float a_l[16][HH];
  __shared__ float we_l[HH][DE_IN];
  __shared__ float be_l[HH];

  int tid  = threadIdx.y * 32 + threadIdx.x;
  int i0   = blockIdx.x * 16;
  int spl  = blockIdx.y;
  int jst  = spl * JCH;
  int wave = threadIdx.y;
  int lane = threadIdx.x;
  int half = lane >> 4;
  int l    = lane & 15;

  if (tid < 64) {
    int ti3 = tid >> 2, h2 = tid & 3;
    m_l[ti3][h2] = NEG_INF;
    s_l[ti3][h2] = 0.f;
    a_l[ti3][h2] = avec[(size_t)(i0 + ti3) * HH + h2];
  }
  if (tid < HH * DE_IN) we_l[tid >> 4][tid & 15] = weff[tid];
  if (tid < HH)         be_l[tid] = beff[tid];

  // per-wave tiles: t = wave*2 + {0,1};  h = t>>2, dtile = t&3
  int t0 = wave * 2, t1 = t0 + 1;
  int h0 = t0 >> 2, dt0 = t0 & 3;
  int h1 = t1 >> 2, dt1 = t1 & 3;
  v8f c0 = {}, c1 = {};

  // logit-work mapping: each thread owns (ti, jb..jb+3), same ti for all 4
  int base = tid << 2;
  int ti = base >> 6;
  int jb = base & 63;
  int irow = i0 + ti;

  for (int j0 = jst; j0 < jst + JCH; j0 += 64) {
    __syncthreads();
    // ---- step 1: logits (with adj mask) ----
    for (int pp = 0; pp < 4; ++pp) {
      int j = jb + pp;
      float av = adj[(size_t)irow * NN + (j0 + j)];
      adjl[ti][j] = av;
      if (j0 + 64 < jst + JCH)  // prefetch next chunk's edge row segment
        __builtin_prefetch(ef + ((size_t)irow * NN + (j0 + 64 + j)) * DE_IN, 0, 1);
      if (av > 0.f) {
        const float4* e4 = (const float4*)(ef + ((size_t)irow * NN + (j0 + j)) * DE_IN);
        float4 ea = e4[0], eb = e4[1], ec = e4[2], ed = e4[3];
        float ev[16] = {ea.x, ea.y, ea.z, ea.w, eb.x, eb.y, eb.z, eb.w,
                        ec.x, ec.y, ec.z, ec.w, ed.x, ed.y, ed.z, ed.w};
        #pragma unroll
        for (int h = 0; h < HH; ++h) {
          float d = be_l[h];
          #pragma unroll
          for (int cc = 0; cc < DE_IN; ++cc) d += ev[cc] * we_l[h][cc];
          Wl[h][ti][j] = a_l[ti][h] + bvec[(size_t)(j0 + j) * HH + h] + d;
        }
      } else {
        #pragma unroll
        for (int h = 0; h < HH; ++h) Wl[h][ti][j] = NEG_INF;
      }
    }
    __syncthreads();
    // ---- step 2: running-max update & accumulator scale factors ----
    if (tid < 64) {
      int ti2 = tid >> 2, h2 = tid & 3;
      float cm = NEG_INF;
      for (int j = 0; j < 64; ++j) cm = fmaxf(cm, Wl[h2][ti2][j]);
      float mo = m_l[ti2][h2];
      float mn = fmaxf(mo, cm);
      float sc = (mo == mn) ? 1.f : __expf(mo - mn);  // handles -inf,-inf -> 1
      m_l[ti2][h2]  = mn;
      sc_l[ti2][h2] = sc;
      s_l[ti2][h2] *= sc;
    }
    __syncthreads();
    // ---- step 3: exp, denominator accumulation, numerator weight = e*adj ----
    {
      float part[HH] = {0.f, 0.f, 0.f, 0.f};
      for (int pp = 0; pp < 4; ++pp) {
        int j = jb + pp;
        float av = adjl[ti][j];
        #pragma unroll
        for (int h = 0; h < HH; ++h) {
          float e = 0.f;
          if (av > 0.f) e = __expf(Wl[h][ti][j] - m_l[ti][h]);
          part[h] += e;
          Wl[h][ti][j] = e * av;
        }
      }
      #pragma unroll
      for (int h = 0; h < HH; ++h)
        if (part[h] > 0.f) atomicAdd(&s_l[ti][h], part[h]);
    }
    __syncthreads();
    // ---- step 4: rescale accumulators, then C += W_chunk x V_chunk (WMMA) ----
    #pragma unroll
    for (int r = 0; r < 8; ++r) {
      c0[r] *= sc_l[r + half * 8][h0];
      c1[r] *= sc_l[r + half * 8][h1];
    }
    int colb0 = h0 * DHH + dt0 * 16 + l;
    int colb1 = h1 * DHH + dt1 * 16 + l;
    for (int kb = 0; kb < 64; kb += 4) {
      int k0 = kb + half * 2;
      v2f a0, b0, a1, b1;
      a0.x = Wl[h0][l][k0 + 0];
      a0.y = Wl[h0][l][k0 + 1];
      b0.x = v[(size_t)(j0 + k0 + 0) * DD + colb0];
      b0.y = v[(size_t)(j0 + k0 + 1) * DD + colb0];
      c0 = wmma_f32(a0, b0, c0);
      a1.x = Wl[h1][l][k0 + 0];
      a1.y = Wl[h1][l][k0 + 1];
      b1.x = v[(size_t)(j0 + k0 + 0) * DD + colb1];
      b1.y = v[(size_t)(j0 + k0 + 1) * DD + colb1];
      c1 = wmma_f32(a1, b1, c1);
    }
  }

  __syncthreads();
  // ---- emit partials: P (unnormalized, at scale m_l), m, s ----
  if (tid < 64) {
    int ti2 = tid >> 2, h2 = tid & 3;
    size_t o = ((size_t)spl * NN + (i0 + ti2)) * HH + h2;
    m_out[o] = m_l[ti2][h2];
    s_out[o] = s_l[ti2][h2];
  }
  int colb0 = h0 * DHH + dt0 * 16 + l;
  int colb1 = h1 * DHH + dt1 * 16 + l;
  #pragma unroll
  for (int r = 0; r < 8; ++r) {
    int row = r + half * 8;
    size_t o = ((size_t)spl * NN + (i0 + row)) * DD;
    Pacc[o + colb0] = c0[r];
    Pacc[o + colb1] = c1[r];
  }
}

// ---------------------------------------------------------------------------
// Combine split partials:  msg = sum_s w_s*P_s / sum_s w_s*s_s,
// w_s = exp(m_s - m_global). One block per receiver row.
// ---------------------------------------------------------------------------
__global__ __launch_bounds__(256) void combine_k(
    const float* __restrict__ m_in, const float* __restrict__ s_in,
    const float* __restrict__ Pacc, float* __restrict__ msg) {
  __shared__ float wsc[JS][HH];
  __shared__ float den[HH];
  int i = blockIdx.x, t = threadIdx.x;
  if (t < HH) {
    float mg = NEG_INF;
    for (int s = 0; s < JS; ++s)
      mg = fmaxf(mg, m_in[((size_t)s * NN + i) * HH + t]);
    float d = 0.f;
    for (int s = 0; s < JS; ++s) {
      float ms = m_in[((size_t)s * NN + i) * HH + t];
      float w = (ms == NEG_INF) ? 0.f : __expf(ms - mg);
      wsc[s][t] = w;
      d += w * s_in[((size_t)s * NN + i) * HH + t];
    }
    den[t] = d;
  }
  __syncthreads();
  int h = t >> 6;                      // 64 cols per head
  float acc = 0.f;
  #pragma unroll
  for (int s = 0; s < JS; ++s)
    acc += wsc[s][h] * Pacc[((size_t)s * NN + i) * DD + t];
  float d = den[h];
  msg[(size_t)i * DD + t] = (d > 0.f) ? acc / d : 0.f;
}

// ---------------------------------------------------------------------------
// Row layernorm over D=256.
// ---------------------------------------------------------------------------
__global__ __launch_bounds__(256) void layernorm_k(const float* __restrict__ x,
                                                   const float* __restrict__ g,
                                                   const float* __restrict__ b,
                                                   float* __restrict__ out) {
  __shared__ float red[256];
  int i = blockIdx.x, t = threadIdx.x;
  float val = x[(size_t)i * DD + t];
  red[t] = val;
  __syncthreads();
  for (int off = 128; off > 0; off >>= 1) {
    if (t < off) red[t] += red[t + off];
    __syncthreads();
  }
  float mu = red[0] * (1.f / DD);
  __syncthreads();
  float dv = val - mu;
  red[t] = dv * dv;
  __syncthreads();
  for (int off = 128; off > 0; off >>= 1) {
    if (t < off) red[t] += red[t + off];
    __syncthreads();
  }
  float var = red[0] * (1.f / DD);
  out[(size_t)i * DD + t] = dv * rsqrtf(var + 1e-5f) * g[t] + b[t];
}

// ---------------------------------------------------------------------------
extern "C" void kernel_launch(void* const* d_in, const int* in_sizes, int n_in,
                              void* d_out, int out_size, void* d_ws, size_t ws_size,
                              hipStream_t stream) {
  (void)in_sizes; (void)n_in; (void)out_size; (void)ws_size;
  const float* h    = (const float*)d_in[0];
  const float* adj  = (const float*)d_in[1];
  const float* ef   = (const float*)d_in[2];
  const float* Wq   = (const float*)d_in[3];
  const float* bq   = (const float*)d_in[4];
  const float* Wk   = (const float*)d_in[5];
  const float* bk   = (const float*)d_in[6];
  const float* Wv   = (const float*)d_in[7];
  const float* bvv  = (const float*)d_in[8];
  const float* Ws   = (const float*)d_in[9];
  const float* bs   = (const float*)d_in[10];
  const float* We   = (const float*)d_in[11];
  const float* be   = (const float*)d_in[12];
  const float* u    = (const float*)d_in[13];
  const float* Wo   = (const float*)d_in[14];
  const float* bo   = (const float*)d_in[15];
  const float* ln_g = (const float*)d_in[16];
  const float* ln_b = (const float*)d_in[17];
  float* out = (float*)d_out;

  const size_t ND = (size_t)NN * DD;
  const size_t NH = (size_t)NN * HH;
  float* ws   = (float*)d_ws;
  float* q    = ws;                 // reused as msg after qk_dots
  float* k    = ws + ND;            // reused as x-buffer after qk_dots
  float* v    = ws + 2 * ND;
  float* sp   = ws + 3 * ND;
  float* Pacc = ws + 4 * ND;        // JS * ND
  float* avec = ws + (4 + JS) * ND;
  float* bvec = avec + NH;
  float* m_p  = bvec + NH;          // JS * NH
  float* s_p  = m_p + JS * NH;      // JS * NH
  float* weff = s_p + JS * NH;
  float* beff = weff + HH * DE_IN;
  float* msg  = q;
  float* xb   = k;

  prep_weff<<<1, 64, 0, stream>>>(u, We, be, weff, beff);

  dim3 gblk(32, 4);
  int gemm_blocks = (NN / 16) * (DD / 16) / 4;   // 384
  gemm_wmma<<<gemm_blocks, gblk, 0, stream>>>(h, Wq, bq, q,  nullptr, nullptr);
  gemm_wmma<<<gemm_blocks, gblk, 0, stream>>>(h, Wk, bk, k,  nullptr, nullptr);
  gemm_wmma<<<gemm_blocks, gblk, 0, stream>>>(h, Wv, bvv, v, nullptr, nullptr);
  gemm_wmma<<<gemm_blocks, gblk, 0, stream>>>(h, Ws, bs, sp, nullptr, nullptr);

  qk_dots<<<(NN * HH + 255) / 256, 256, 0, stream>>>(q, k, u, avec, bvec);

  attn_split<<<dim3(NN / 16, JS), dim3(32, 8), 0, stream>>>(
      adj, ef, v, avec, bvec, weff, beff, Pacc, m_p, s_p);

  combine_k<<<NN, 256, 0, stream>>>(m_p, s_p, Pacc, msg);

  gemm_wmma<<<gemm_blocks, gblk, 0, stream>>>(msg, Wo, bo, xb, h, sp);

  layernorm_k<<<NN, 256, 0, stream>>>(xb, ln_g, ln_b, out);
}